// Arm_82119774699744
// MI455X (gfx1250) — compile-verified
//
#include <hip/hip_runtime.h>

typedef __attribute__((ext_vector_type(2))) float v2f;
typedef __attribute__((ext_vector_type(8))) float v8f;

#define KW 15
#define BN_EPS 1e-5f

// section sizes (floats)
#define A1_ELEMS (8 * 72 * 987)       // conv1 act, stride 987
#define P1_SEC   (8 * 72 * 336)       // pooled1 per phase, stride 336 (width <= 329)
#define A2_SEC   (8 * 144 * 320)      // conv2 act per phase, stride 320 (width <= 315)
#define P2_SEC   (8 * 144 * 112)      // pooled2 per combo, stride 112 (width <= 105)
#define A3_SEC   (8 * 288 * 96)       // conv3 act per combo, stride 96 (width <= 91)
#define P3_SEC   (8 * 288 * 32)       // pooled3 per combo, stride 32 (width <= 30)
#define SLOT_ELEMS 1990656            // max(A1, A2, A3) == max(P1, P2, P3) == 9*8*288*96

// ---------------------------------------------------------------------------
// AvgPool3 with phase offset: dst[sec][bc][j] = mean3(src[secSrc][bc][phase+3j])
// ---------------------------------------------------------------------------
template <int MODE>
__global__ void __launch_bounds__(256)
pool3x(const float* __restrict__ src, float* __restrict__ dst)
{
    const int sec = blockIdx.z;
    const int idx = blockIdx.x * 256 + threadIdx.x;

    int C, srcStride, dstStride, W, phase;
    const float* sbase;
    if constexpr (MODE == 1) {            // a1 -> pooled1, sec = p1
        C = 72; srcStride = 987; dstStride = 336;
        W = (sec == 0) ? 329 : 328;
        phase = sec; sbase = src;
    } else if constexpr (MODE == 2) {     // a2 -> pooled2, sec = q = p1+3*p2
        C = 144; srcStride = 320; dstStride = 112;
        W = (sec == 0) ? 105 : 104;
        const int p1 = sec % 3; phase = sec / 3;
        sbase = src + (size_t)p1 * A2_SEC;
    } else {                              // a3 -> pooled3, sec = r = q+9*p3
        C = 288; srcStride = 96; dstStride = 32;
        const int q = sec % 9; const int p3 = sec / 9;
        W = (p3 == 0) ? 30 : ((p3 == 1 && q == 0) ? 30 : 29);
        phase = p3; sbase = src + (size_t)q * A3_SEC;
    }

    if (idx >= 8 * C * dstStride) return;
    const int bc = idx / dstStride;
    const int j  = idx - bc * dstStride;
    if (j >= W) return;
    const float* p = sbase + (size_t)bc * srcStride + phase + 3 * j;
    float* dbase;
    if constexpr (MODE == 1) dbase = dst + (size_t)sec * P1_SEC;
    else if constexpr (MODE == 2) dbase = dst + (size_t)sec * P2_SEC;
    else dbase = dst + (size_t)sec * P3_SEC;
    dbase[idx] = (p[0] + p[1] + p[2]) * (1.0f / 3.0f);
}

// ---------------------------------------------------------------------------
// Conv(valid,[1x15]) + BN + ELU as WMMA GEMM. One wave computes a 16x64 output
// macro-tile (1 M-tile x 4 N-tiles) so each b64 weight load feeds 4 WMMAs.
//   D[co, (b,u)] = sum_{ci,kk} W[co, ci*15+kk] * In[b][ci][u+kk]
// ---------------------------------------------------------------------------
template <int MODE, int O, int Ci>
__global__ void __launch_bounds__(256)
conv_bn_elu(const float* __restrict__ src, const float* __restrict__ w,
            const float* __restrict__ bias, const float* __restrict__ gamma,
            const float* __restrict__ beta, const float* __restrict__ mean,
            const float* __restrict__ var, float* __restrict__ dst)
{
    constexpr int K = Ci * KW;
    constexpr int Mtiles = (O + 15) / 16;

    const int lane = threadIdx.x & 31;
    const int wv   = threadIdx.x >> 5;
    const int half = lane >> 4;   // K-pair select (lanes 16-31 hold K=2,3)
    const int lidx = lane & 15;   // row (A) / col (B) index

    int Wout, inStride, outStride;
    const float* in;
    float* out;
    if constexpr (MODE == 1) {                     // x -> a1
        Wout = 987; inStride = 1001; outStride = 987;
        in = src; out = dst;
    } else if constexpr (MODE == 2) {              // pooled1[p1] -> a2[p1]
        const int p1 = blockIdx.z;
        Wout = (p1 == 0) ? 315 : 314;
        inStride = 336; outStride = 320;
        in = src + (size_t)p1 * P1_SEC; out = dst + (size_t)p1 * A2_SEC;
    } else {                                       // pooled2[q] -> a3[q]
        const int q = blockIdx.z;
        Wout = (q == 0) ? 91 : 90;
        inStride = 112; outStride = 96;
        in = src + (size_t)q * P2_SEC; out = dst + (size_t)q * A3_SEC;
    }

    const int Ntot = 8 * Wout;
    const int NtM  = (Ntot + 63) >> 6;             // 64-wide macro tiles
    const int tile = blockIdx.x * 8 + wv;          // wave-uniform
    if (tile >= Mtiles * NtM) return;
    const int mt  = tile / NtM;
    const int ntm = tile - mt * NtM;

    // per-column state for the 4 N-tiles (clamped so loads stay in bounds and
    // EXEC stays all-ones through the WMMA loop)
    int nv[4], bArr[4], uArr[4];
    const float* bases[4];
#pragma unroll
    for (int j = 0; j < 4; ++j) {
        const int n    = ntm * 64 + j * 16 + lidx;
        const int nEff = (n < Ntot) ? n : (Ntot - 1);
        const int b    = nEff / Wout;
        const int u    = nEff - b * Wout;
        nv[j] = n; bArr[j] = b; uArr[j] = u;
        bases[j] = in + (size_t)b * Ci * inStride + u;
    }

    const int mA    = mt * 16 + lidx;
    const int mAeff = (mA < O) ? mA : (O - 1);
    const float* wrow = w + (size_t)mAeff * K;

    v8f acc[4] = {};
    for (int k0 = 0; k0 < K; k0 += 4) {
        const int ks  = k0 + 2 * half;             // even -> 8B-aligned A pair
        const v2f A   = *(const v2f*)(wrow + ks);
        const int ci0 = ks / KW;
        const int kk0 = ks - ci0 * KW;
        const int ci1 = (ks + 1) / KW;
        const int kk1 = (ks + 1) - ci1 * KW;
        const int off0 = ci0 * inStride + kk0;     // shared across the 4 tiles
        const int off1 = ci1 * inStride + kk1;
#pragma unroll
        for (int j = 0; j < 4; ++j) {
            v2f Bv;
            Bv.x = bases[j][off0];
            Bv.y = bases[j][off1];
            acc[j] = __builtin_amdgcn_wmma_f32_16x16x4_f32(
                false, A, false, Bv, (short)0, acc[j], false, false);
        }
    }

    // epilogue: BN folded to y = acc*sc + off, then ELU
#pragma unroll
    for (int v = 0; v < 8; ++v) {
        const int co  = mt * 16 + v + 8 * half;    // C/D row = v + 8*half
        const bool cok = (co < O);
        const int coc = cok ? co : (O - 1);
        const float sc  = gamma[coc] * rsqrtf(var[coc] + BN_EPS);
        const float off = (bias[coc] - mean[coc]) * sc + beta[coc];
#pragma unroll
        for (int j = 0; j < 4; ++j) {
            if (cok && nv[j] < Ntot) {
                float y = acc[j][v] * sc + off;
                y = (y > 0.0f) ? y : expm1f(y);    // ELU(alpha=1)
                out[((size_t)bArr[j] * O + co) * outStride + uArr[j]] = y;
            }
        }
    }
}

// ---------------------------------------------------------------------------
// FC over flattened (288 x 11) pooled3 features, accumulated over all 501
// crops: crop s -> combo r = s%27 (base-3 digits = pool phases), start m0=s/27.
// One block per (batch,out); FC weight row staged in LDS.
// ---------------------------------------------------------------------------
__global__ void __launch_bounds__(256)
fc_accum(const float* __restrict__ pooled3, const float* __restrict__ wfc,
         const float* __restrict__ bfc, float* __restrict__ out)
{
    __shared__ float wsh[3168];
    __shared__ float sdata[256];
    const int b = blockIdx.x >> 2;
    const int o = blockIdx.x & 3;
    for (int f = threadIdx.x; f < 3168; f += 256)
        wsh[f] = wfc[o * 3168 + f];
    __syncthreads();

    float acc = 0.0f;
    for (int i = 0; i < 501; ++i) {
        const int s  = (i == 0) ? 0 : (i + 1);     // shifts: {0, 2..501}
        const int r  = s % 27;
        const int m0 = s / 27;
        const float* base = pooled3 + ((size_t)r * 8 + b) * (288 * 32) + m0;
        for (int f = threadIdx.x; f < 3168; f += 256) {
            const int c = f / 11;
            const int t = f - c * 11;
            acc += wsh[f] * base[c * 32 + t];
        }
    }
    sdata[threadIdx.x] = acc;
    __syncthreads();
    for (int st = 128; st > 0; st >>= 1) {
        if (threadIdx.x < st) sdata[threadIdx.x] += sdata[threadIdx.x + st];
        __syncthreads();
    }
    if (threadIdx.x == 0)
        out[blockIdx.x] = bfc[o] + sdata[0] * (1.0f / 501.0f);
}

extern "C" void kernel_launch(void* const* d_in, const int* in_sizes, int n_in,
                              void* d_out, int out_size, void* d_ws, size_t ws_size,
                              hipStream_t stream)
{
    (void)in_sizes; (void)n_in; (void)out_size; (void)ws_size;

    const float* x   = (const float*)d_in[0];
    const float* w1  = (const float*)d_in[1];
    const float* b1  = (const float*)d_in[2];
    const float* g1  = (const float*)d_in[3];
    const float* be1 = (const float*)d_in[4];
    const float* m1  = (const float*)d_in[5];
    const float* v1  = (const float*)d_in[6];
    const float* w2  = (const float*)d_in[7];
    const float* b2  = (const float*)d_in[8];
    const float* g2  = (const float*)d_in[9];
    const float* be2 = (const float*)d_in[10];
    const float* m2  = (const float*)d_in[11];
    const float* v2  = (const float*)d_in[12];
    const float* w3  = (const float*)d_in[13];
    const float* b3  = (const float*)d_in[14];
    const float* g3  = (const float*)d_in[15];
    const float* be3 = (const float*)d_in[16];
    const float* m3  = (const float*)d_in[17];
    const float* v3  = (const float*)d_in[18];
    const float* wfc = (const float*)d_in[19];
    const float* bfc = (const float*)d_in[20];

    // two ping-pong slots; lifetimes alternate: a1->p1->a2->p2->a3->p3
    float* slotA = (float*)d_ws;                 // a1, a2, a3
    float* slotB = slotA + SLOT_ELEMS;           // pooled1, pooled2, pooled3
    // total workspace: 2 * 1,990,656 floats = 15.9 MB

    // conv1: M=72 (5 tiles), N=7896 -> 124 macro tiles -> 620 waves -> 78 blocks
    conv_bn_elu<1, 72, 36><<<dim3(78, 1, 1), 256, 0, stream>>>(
        x, w1, b1, g1, be1, m1, v1, slotA);
    // pool1: 3 phases
    pool3x<1><<<dim3(756, 1, 3), 256, 0, stream>>>(slotA, slotB);
    // conv2: per p1: M=144 (9 tiles), N<=2520 -> 40 macro -> 360 waves -> 45 blocks
    conv_bn_elu<2, 144, 72><<<dim3(45, 1, 3), 256, 0, stream>>>(
        slotB, w2, b2, g2, be2, m2, v2, slotA);
    // pool2: 9 combos
    pool3x<2><<<dim3(504, 1, 9), 256, 0, stream>>>(slotA, slotB);
    // conv3: per q: M=288 (18 tiles), N<=728 -> 12 macro -> 216 waves -> 27 blocks
    conv_bn_elu<3, 288, 144><<<dim3(27, 1, 9), 256, 0, stream>>>(
        slotB, w3, b3, g3, be3, m3, v3, slotA);
    // pool3: 27 combos
    pool3x<3><<<dim3(288, 1, 27), 256, 0, stream>>>(slotA, slotB);
    // FC + crop accumulation
    fc_accum<<<dim3(32, 1, 1), 256, 0, stream>>>(slotB, wfc, bfc, (float*)d_out);
}